// GIN_26465588478351
// MI455X (gfx1250) — compile-verified
//
#include <hip/hip_runtime.h>

// ---------------------------------------------------------------------------
// GIN forward (50000 nodes, 800000 edges, nfeat=128, nhid=64, 3 layers)
// fp32 end-to-end; GEMMs on V_WMMA_F32_16X16X4_F32 (wave32, 16x16 tile/wave).
// ---------------------------------------------------------------------------

#define NFEAT  128
#define NHID   64
#define NLAYER 3
#define BN_EPS 1e-5f

typedef __attribute__((ext_vector_type(2))) float v2f;
typedef __attribute__((ext_vector_type(8))) float v8f;

// ---------------------------------------------------------------------------
// C[M x N] = act(A[M x K] @ W[K x N] + bias)   (N multiple of 16, K mult of 4)
// One wave (32 threads) per 16x16 output tile. grid = (ceil(M/16), N/16).
//
// Fragment layouts per CDNA5 ISA 7.12.2 (32-bit, wave32):
//   A 16x4 : lanes 0-15 -> M=lane,   v0=K0, v1=K1 ; lanes 16-31 -> v0=K2, v1=K3
//   B 4x16 : v0: lanes0-15 K=0, lanes16-31 K=2 ; v1: K=1 / K=3 ; col = lane%16
//   C/D    : VGPR r: lanes0-15 -> (M=r, N=lane), lanes16-31 -> (M=r+8, N=lane-16)
// ---------------------------------------------------------------------------
__global__ void __launch_bounds__(32)
gin_gemm_wmma_f32(const float* __restrict__ A, const float* __restrict__ W,
                  const float* __restrict__ bias, float* __restrict__ C,
                  int M, int K, int N, int do_relu)
{
    const int tileM = blockIdx.x * 16;
    const int tileN = blockIdx.y * 16;
    const int lane  = threadIdx.x;       // 0..31
    const int half  = lane >> 4;         // 0 | 1
    const int l16   = lane & 15;

    const int arow  = tileM + l16;
    const int arow_c = arow < M ? arow : (M - 1);   // clamp (M is 16-aligned here anyway)
    const float* Arow = A + (size_t)arow_c * K;

    v8f acc = {};
    for (int k0 = 0; k0 < K; k0 += 4) {
        const int ka = k0 + 2 * half;          // K index for A/B vgpr0 in this lane-half
        v2f a, b;
        a.x = Arow[ka];
        a.y = Arow[ka + 1];
        b.x = W[(size_t)ka * N + tileN + l16];
        b.y = W[(size_t)(ka + 1) * N + tileN + l16];
        // 8 args: (neg_a, A, neg_b, B, c_mod, C, reuse_a, reuse_b)
        acc = __builtin_amdgcn_wmma_f32_16x16x4_f32(
                false, a, false, b, (short)0, acc, false, false);
    }

    const int col = tileN + l16;
    const float bv = bias ? bias[col] : 0.0f;
#pragma unroll
    for (int r = 0; r < 8; ++r) {
        const int row = tileM + r + 8 * half;
        if (row < M) {
            float v = acc[r] + bv;
            if (do_relu) v = v > 0.0f ? v : 0.0f;
            C[(size_t)row * N + col] = v;
        }
    }
}

// ---------------------------------------------------------------------------
// agg = h   (float4 copy)
// ---------------------------------------------------------------------------
__global__ void gin_copy4(const float4* __restrict__ src, float4* __restrict__ dst,
                          int n4)
{
    int i = blockIdx.x * blockDim.x + threadIdx.x;
    if (i < n4) dst[i] = src[i];
}

// ---------------------------------------------------------------------------
// agg[dst] += h[src] over edges. 16 lanes per edge, float4 gather per lane,
// 4x global_atomic_add_f32 per lane (no-return atomics -> STOREcnt path).
// ---------------------------------------------------------------------------
__global__ void gin_scatter_add(const long long* __restrict__ ei,
                                const float* __restrict__ h,
                                float* __restrict__ agg, int n_edges)
{
    int idx = blockIdx.x * blockDim.x + threadIdx.x;
    if (idx >= n_edges * 16) return;
    const int e  = idx >> 4;         // edge id
    const int c4 = idx & 15;         // float4 chunk within the 64-wide feature row
    const int src = (int)ei[e];
    const int dst = (int)ei[n_edges + e];
    const float4 v = *(const float4*)(h + (size_t)src * NHID + c4 * 4);
    float* p = agg + (size_t)dst * NHID + c4 * 4;
    atomicAdd(p + 0, v.x);
    atomicAdd(p + 1, v.y);
    atomicAdd(p + 2, v.z);
    atomicAdd(p + 3, v.w);
}

// ---------------------------------------------------------------------------
// BatchNorm: column sums / sums-of-squares (64 columns).
// Block = 256 threads = 64 cols x 4 row-groups, 128 rows per block,
// LDS cross-group reduce, then 128 float atomics per block (64 to sums, 64 to sumsq).
// ---------------------------------------------------------------------------
__global__ void __launch_bounds__(256)
gin_bn_stats(const float* __restrict__ h, int n_rows,
             float* __restrict__ sums, float* __restrict__ sumsq)
{
    __shared__ float s1[256];
    __shared__ float s2[256];
    const int c = threadIdx.x & 63;
    const int g = threadIdx.x >> 6;              // 0..3
    const int row0 = blockIdx.x * 128;
    float a = 0.0f, b = 0.0f;
    for (int i = g; i < 128; i += 4) {
        const int r = row0 + i;
        if (r < n_rows) {
            const float v = h[(size_t)r * NHID + c];
            a += v;
            b += v * v;
        }
    }
    s1[threadIdx.x] = a;
    s2[threadIdx.x] = b;
    __syncthreads();
    if (g == 0) {
        a = s1[c] + s1[c + 64] + s1[c + 128] + s1[c + 192];
        b = s2[c] + s2[c + 64] + s2[c + 128] + s2[c + 192];
        atomicAdd(&sums[c],  a);
        atomicAdd(&sumsq[c], b);
    }
}

__global__ void gin_zero_stats(float* __restrict__ p)   // zero 128 floats
{
    if (threadIdx.x < 128) p[threadIdx.x] = 0.0f;
}

// ---------------------------------------------------------------------------
// out = gamma*(in - mu)*rsqrt(var + eps) + beta, mu/var from sums/sumsq.
// ---------------------------------------------------------------------------
__global__ void gin_bn_apply(const float* __restrict__ in, float* __restrict__ out,
                             const float* __restrict__ sums,
                             const float* __restrict__ sumsq,
                             const float* __restrict__ gamma,
                             const float* __restrict__ beta,
                             int n_elems, float inv_n)
{
    int i = blockIdx.x * blockDim.x + threadIdx.x;
    if (i >= n_elems) return;
    const int c = i & (NHID - 1);
    const float mu  = sums[c] * inv_n;
    const float var = sumsq[c] * inv_n - mu * mu;
    out[i] = gamma[c] * (in[i] - mu) * rsqrtf(var + BN_EPS) + beta[c];
}

// ---------------------------------------------------------------------------
// Host-side launch sequence (graph-capturable: only kernel launches on stream)
// ---------------------------------------------------------------------------
extern "C" void kernel_launch(void* const* d_in, const int* in_sizes, int n_in,
                              void* d_out, int out_size, void* d_ws, size_t ws_size,
                              hipStream_t stream)
{
    const float*     x     = (const float*)d_in[0];
    const long long* ei    = (const long long*)d_in[1];
    const float*     W_t   = (const float*)d_in[2];
    const float*     b_t   = (const float*)d_in[3];
    const float*     g_t   = (const float*)d_in[4];
    const float*     bb_t  = (const float*)d_in[5];
    const float*     W1    = (const float*)d_in[6];
    const float*     W2    = (const float*)d_in[7];
    const float*     gamma = (const float*)d_in[8];
    const float*     beta  = (const float*)d_in[9];
    float*           out   = (float*)d_out;

    const int M = in_sizes[0] / NFEAT;   // 50000 nodes
    const int E = in_sizes[1] / 2;       // 800000 edges
    const int elems = M * NHID;
    const float inv_n = 1.0f / (float)M;

    // workspace layout
    const size_t stride = (size_t)M * NHID;
    float* H     = (float*)d_ws;
    float* TMP   = H + stride;
    float* AGG   = TMP + stride;
    float* SUMS  = AGG + stride;
    float* SUMSQ = SUMS + NHID;

    dim3 gemm_grid((M + 15) / 16, NHID / 16);
    const int bn_blocks  = (M + 127) / 128;
    const int ew_blocks  = (elems + 255) / 256;
    const int cp_blocks  = (elems / 4 + 255) / 256;
    const int sc_blocks  = (E * 16 + 255) / 256;

    // input transform: H = x @ W_t + b_t, then BN(g_t, bb_t)
    gin_gemm_wmma_f32<<<gemm_grid, 32, 0, stream>>>(x, W_t, b_t, H, M, NFEAT, NHID, 0);
    gin_zero_stats<<<1, 128, 0, stream>>>(SUMS);
    gin_bn_stats<<<bn_blocks, 256, 0, stream>>>(H, M, SUMS, SUMSQ);
    gin_bn_apply<<<ew_blocks, 256, 0, stream>>>(H, H, SUMS, SUMSQ, g_t, bb_t, elems, inv_n);

    for (int l = 0; l < NLAYER; ++l) {
        // agg = H + segment_sum(H[src] -> dst)
        gin_copy4<<<cp_blocks, 256, 0, stream>>>((const float4*)H, (float4*)AGG, elems / 4);
        gin_scatter_add<<<sc_blocks, 256, 0, stream>>>(ei, H, AGG, E);

        // TMP = relu(AGG @ W1[l]); AGG = relu(TMP @ W2[l])
        gin_gemm_wmma_f32<<<gemm_grid, 32, 0, stream>>>(
            AGG, W1 + (size_t)l * NHID * NHID, nullptr, TMP, M, NHID, NHID, 1);
        gin_gemm_wmma_f32<<<gemm_grid, 32, 0, stream>>>(
            TMP, W2 + (size_t)l * NHID * NHID, nullptr, AGG, M, NHID, NHID, 1);

        // BN
        gin_zero_stats<<<1, 128, 0, stream>>>(SUMS);
        gin_bn_stats<<<bn_blocks, 256, 0, stream>>>(AGG, M, SUMS, SUMSQ);
        float* dst = (l == NLAYER - 1) ? out : H;
        gin_bn_apply<<<ew_blocks, 256, 0, stream>>>(
            AGG, dst, SUMS, SUMSQ, gamma + (size_t)l * NHID, beta + (size_t)l * NHID,
            elems, inv_n);
    }
}